// QuantizedEmbedding_2731599200973
// MI455X (gfx1250) — compile-verified
//
#include <hip/hip_runtime.h>
#include <hip/hip_bf16.h>

// Quantized int4 embedding gather + dequant for MI455X (gfx1250).
//
// Shapes (from reference):
//   indices : [4, 4096]            int   (16384 tokens)
//   weight  : [128000, 1024]       int8  (two int4 nibbles per byte -> 2048 cols)
//   scales  : [128000, 8]          fp32  (group size 256)
//   out     : [4, 4096, 2048]      fp32
//
// Pure bandwidth problem: ~128 MB of output stores + ~16.5 MB of reads
// => ~6.2 us floor at 23.3 TB/s. No matrix contraction exists, so WMMA is
// deliberately not used; instead we optimize the data-movement path:
//   - fully coalesced 128-bit stores (lane -> 4 contiguous output columns)
//   - non-temporal store hints so the streaming output does not evict the
//     weight table from the 192 MB L2 (the whole packed table is ~128 MB)
//   - wave-uniform index -> scalar load of the gather base

#define PACKED_COLS 1024   // bytes per row
#define EMB_DIM     2048   // output floats per row
#define NGROUPS     8      // scales per row (group size 256 columns)

// Native clang vector type: required by __builtin_nontemporal_store
// (HIP's float4 is a class and is rejected by the builtin).
typedef float v4f __attribute__((ext_vector_type(4)));

struct FPair { float lo, hi; };

__device__ __forceinline__ FPair dequant_byte(int p, float s) {
    // Faithful to torchchat / reference:
    //   even nibble: trunc(p / 16)   (C integer division truncates)
    //   odd  nibble: p mod 16 (floor) == p & 15 for two's complement
    // both minus 8, times group scale.
    FPair r;
    r.lo = (float)((p / 16) - 8) * s;
    r.hi = (float)((p & 15) - 8) * s;
    return r;
}

__global__ void __launch_bounds__(256)
qemb4_dequant_kernel(const int* __restrict__ indices,
                     const signed char* __restrict__ weight,
                     const float* __restrict__ scales,
                     float* __restrict__ out)
{
    const int token = blockIdx.x;          // one block per gathered row
    const int t     = threadIdx.x;         // 0..255

    // Wave-uniform gather index -> scalar load + SGPR base address.
    const long long row = (long long)indices[token];

    const signed char* __restrict__ wrow = weight + row * (long long)PACKED_COLS;
    const float*       __restrict__ srow = scales + row * (long long)NGROUPS;
    float*             __restrict__ orow = out + (long long)token * EMB_DIM;

    // Column remap for perfect store coalescing:
    //   segment k (k=0,1), thread t produces output columns [1024k + 4t, +4),
    //   which come from packed bytes {512k + 2t, 512k + 2t + 1}.
    // A float4-wide store across a wave is then a dense 512 B span; the
    // block's two segments cover the full 8 KB row contiguously.
#pragma unroll
    for (int k = 0; k < 2; ++k) {
        const int byte_off = 512 * k + 2 * t;
        const int col      = 1024 * k + 4 * t;

        const float s = srow[col >> 8];    // group = col / 256 (uniform per wave)

        const unsigned short us =
            *(const unsigned short*)(wrow + byte_off);
        const int b0 = (int)(signed char)(us & 0xFF);
        const int b1 = (int)(signed char)(us >> 8);

        const FPair p0 = dequant_byte(b0, s);
        const FPair p1 = dequant_byte(b1, s);

        v4f v;
        v.x = p0.lo;
        v.y = p0.hi;
        v.z = p1.lo;
        v.w = p1.hi;

        // Streaming output: non-temporal so the 128 MB result does not
        // thrash L2; keeps the weight table L2-resident across replays.
        __builtin_nontemporal_store(v, (v4f*)(orow + col));
    }
}

extern "C" void kernel_launch(void* const* d_in, const int* in_sizes, int n_in,
                              void* d_out, int out_size, void* d_ws, size_t ws_size,
                              hipStream_t stream) {
    const int*         indices = (const int*)d_in[0];
    const signed char* weight  = (const signed char*)d_in[1];
    const float*       scales  = (const float*)d_in[2];
    float*             out     = (float*)d_out;

    const int n_tokens = in_sizes[0];      // 16384 for the reference shapes

    dim3 grid(n_tokens);
    dim3 block(256);
    qemb4_dequant_kernel<<<grid, block, 0, stream>>>(indices, weight, scales, out);
}